// TextDiseasePredictor_47150150975617
// MI455X (gfx1250) — compile-verified
//
#include <hip/hip_runtime.h>
#include <hip/hip_bf16.h>

typedef float v2f __attribute__((ext_vector_type(2)));
typedef float v8f __attribute__((ext_vector_type(8)));
typedef unsigned int u32x4 __attribute__((ext_vector_type(4)));
typedef int i32x4 __attribute__((ext_vector_type(4)));
typedef int i32x8 __attribute__((ext_vector_type(8)));

#define DIM   112
#define SEQ   512
#define NTOK  (16 * 512)          // B*S = 8192
#define NB8   8
#define SPW   896                 // DIM * NB8

// TDM path is compiled unconditionally in the DEVICE pass: if this round
// compiles, tensor_load_to_lds is guaranteed present in the gfx1250 binary.
// (Host pass takes the no-op branch since amdgcn builtins don't exist on x86.)
#if defined(__HIP_DEVICE_COMPILE__)
#define HAS_TDM 1
#endif

// ---------------------------------------------------------------------------
// TDM: DMA a 2D tile (tile_d0 x tile_d1 f32 elements, row stride stride0)
// from global into LDS (dense row-major).  Descriptor per
// CDNA5 ISA.  Issue from ONE wave, then
// s_wait_tensorcnt 0 in that wave before the workgroup barrier.
// ---------------------------------------------------------------------------
#if defined(HAS_TDM)
__device__ __forceinline__ void tdm_load_2d(const void* gaddr, unsigned lds_off,
                                            unsigned tensor_d0, unsigned tensor_d1,
                                            unsigned tile_d0, unsigned tile_d1,
                                            unsigned stride0_elems) {
  u32x4 g0;
  unsigned long long ga = (unsigned long long)(size_t)gaddr;
  g0.x = 1u;                                          // count=1, user descriptor
  g0.y = lds_off;                                     // LDS byte address
  g0.z = (unsigned)(ga & 0xFFFFFFFFu);                // global_addr[31:0]
  g0.w = (unsigned)((ga >> 32) & 0x1FFFFFFu) | 0x80000000u;  // [56:32] | type=2

  i32x8 g1;
  g1[0] = 0x00020000;                                 // data_size = 4 bytes
  g1[1] = (int)((tensor_d0 & 0xFFFFu) << 16);         // tensor_dim0[15:0]
  g1[2] = (int)(((tensor_d0 >> 16) & 0xFFFFu) |
                ((tensor_d1 & 0xFFFFu) << 16));       // dim0[31:16] | dim1[15:0]
  g1[3] = (int)(((tensor_d1 >> 16) & 0xFFFFu) |
                ((tile_d0 & 0xFFFFu) << 16));         // dim1[31:16] | tile_dim0
  g1[4] = (int)(tile_d1 & 0xFFFFu);                   // tile_dim1 (tile_dim2=0)
  g1[5] = (int)stride0_elems;                         // tensor_dim0_stride[31:0]
  g1[6] = 0;
  g1[7] = 0;

  i32x4 z4 = {0, 0, 0, 0};
#if __clang_major__ >= 23
  i32x8 z8 = {0, 0, 0, 0, 0, 0, 0, 0};
  __builtin_amdgcn_tensor_load_to_lds(g0, g1, z4, z4, z8, 0);
#else
  __builtin_amdgcn_tensor_load_to_lds(g0, g1, z4, z4, 0);
#endif
}

__device__ __forceinline__ void tdm_wait() {
  __builtin_amdgcn_s_wait_tensorcnt(0);
}
#endif

// ---------------------------------------------------------------------------
// WMMA fragment helpers (CDNA5 ISA 7.12.2 layouts, wave32).
// ---------------------------------------------------------------------------
__device__ __forceinline__ v2f load_frag(const float* __restrict__ base, int ld) {
  int lane = threadIdx.x & 31;
  int half = lane >> 4, l = lane & 15;
  const float* p = base + (size_t)l * ld + 2 * half;
  v2f f; f.x = p[0]; f.y = p[1];
  return f;
}

__device__ __forceinline__ v8f wmma4(v2f a, v2f b, v8f c) {
  return __builtin_amdgcn_wmma_f32_16x16x4_f32(false, a, false, b,
                                               (short)0, c, false, false);
}

// ---------------------------------------------------------------------------
// Fused QKV projection: one block per 16-row strip; TDM stages the A strip
// once; each of 7 waves computes its 16-col tile of Q, K and V (84 WMMAs),
// sharing every A fragment across the three weight matrices.
// ---------------------------------------------------------------------------
__global__ void qkv_kernel(const float* __restrict__ h,
                           const float* __restrict__ qw, const float* __restrict__ qb,
                           const float* __restrict__ kw, const float* __restrict__ kb,
                           const float* __restrict__ vw, const float* __restrict__ vb,
                           float* __restrict__ q, float* __restrict__ k,
                           float* __restrict__ v) {
  __shared__ float alds[16 * DIM];  // 7 KB A strip
  int mt = blockIdx.x;
  int tid = threadIdx.x;
  int wave = tid >> 5, lane = tid & 31;
  int half = lane >> 4, l = lane & 15;

  const float* Ag = h + ((size_t)mt * 16) * DIM;
#if defined(HAS_TDM)
  if (wave == 0) {
    tdm_load_2d(Ag, (unsigned)(size_t)(void*)alds, DIM, NTOK, DIM, 16, DIM);
    tdm_wait();
  }
#else
  for (int i = tid; i < 16 * DIM; i += 256) alds[i] = Ag[i];
#endif
  __syncthreads();

  if (wave < 7) {
    int n0 = wave * 16;
    const float* Bq = qw + (size_t)n0 * DIM;
    const float* Bk = kw + (size_t)n0 * DIM;
    const float* Bv = vw + (size_t)n0 * DIM;
    v8f cq = {}, ck = {}, cv = {};
#pragma unroll
    for (int kc = 0; kc < 28; ++kc) {
      v2f a  = load_frag(alds + kc * 4, DIM);
      v2f bq = load_frag(Bq + kc * 4, DIM);
      v2f bk = load_frag(Bk + kc * 4, DIM);
      v2f bv = load_frag(Bv + kc * 4, DIM);
      cq = wmma4(a, bq, cq);
      ck = wmma4(a, bk, ck);
      cv = wmma4(a, bv, cv);
    }
    int col = n0 + l;
    float bvq = qb[col], bvk = kb[col], bvv = vb[col];
#pragma unroll
    for (int j = 0; j < 8; ++j) {
      int m = j + 8 * half;
      size_t o = ((size_t)mt * 16 + m) * DIM + col;
      q[o] = (cq[j] + bvq) * 0.5f;   // 1/sqrt(HD=4)
      k[o] = ck[j] + bvk;
      v[o] = cv[j] + bvv;
    }
  }
}

// ---------------------------------------------------------------------------
// Attention: per (qtile, head, batch) wave.  TDM packs the strided K/V head
// slices (512x4, stride 112) densely into LDS; 32 WMMAs produce the 16x512
// scores; masked softmax; ctx from LDS V.
// ---------------------------------------------------------------------------
__global__ void attn_kernel(const float* __restrict__ q,
                            const float* __restrict__ k,
                            const float* __restrict__ v,
                            const int* __restrict__ amask,
                            float* __restrict__ ctx) {
  __shared__ float sc[16 * SEQ];    // 32 KB
  __shared__ float kl[SEQ * 4];     // 8 KB
  __shared__ float vl[SEQ * 4];     // 8 KB
  __shared__ float rinv[16];

  int qt = blockIdx.x, hh = blockIdx.y, b = blockIdx.z;
  int lane = threadIdx.x & 31;
  int half = lane >> 4, l = lane & 15;

  const float* kbase = k + ((size_t)b * SEQ) * DIM + hh * 4;
  const float* vbase = v + ((size_t)b * SEQ) * DIM + hh * 4;

#if defined(HAS_TDM)
  tdm_load_2d(kbase, (unsigned)(size_t)(void*)kl, DIM, SEQ, 4, SEQ, DIM);
  tdm_load_2d(vbase, (unsigned)(size_t)(void*)vl, DIM, SEQ, 4, SEQ, DIM);
  tdm_wait();
#else
  for (int i = lane; i < SEQ * 4; i += 32) {
    int kk = i >> 2, d = i & 3;
    kl[i] = kbase[(size_t)kk * DIM + d];
    vl[i] = vbase[(size_t)kk * DIM + d];
  }
#endif
  __syncthreads();

  // A fragment: Q tile 16x4, loaded once, reused for all 32 key tiles
  const float* qbase = q + ((size_t)(b * SEQ + qt * 16)) * DIM + hh * 4;
  v2f a;
  a.x = qbase[(size_t)l * DIM + 2 * half];
  a.y = qbase[(size_t)l * DIM + 2 * half + 1];

  for (int kt = 0; kt < 32; ++kt) {
    const float* bp = &kl[(kt * 16 + l) * 4 + 2 * half];
    v2f bb; bb.x = bp[0]; bb.y = bp[1];
    v8f c = {};
    c = wmma4(a, bb, c);
    int key = kt * 16 + l;
    int mv = amask[b * SEQ + key];
#pragma unroll
    for (int j = 0; j < 8; ++j) {
      int qr = j + 8 * half;
      sc[qr * SEQ + key] = (mv == 0) ? -1.0e9f : c[j];
    }
  }
  __syncthreads();

  // softmax: row r = lane&15, each half-wave covers 256 keys
  {
    int r = l;
    float mx = -3.4e38f;
    int k0 = half * 256;
    for (int kk = k0; kk < k0 + 256; ++kk) mx = fmaxf(mx, sc[r * SEQ + kk]);
    mx = fmaxf(mx, __shfl_xor(mx, 16, 32));
    float sum = 0.0f;
    for (int kk = k0; kk < k0 + 256; ++kk) {
      float e = __expf(sc[r * SEQ + kk] - mx);
      sc[r * SEQ + kk] = e;
      sum += e;
    }
    sum += __shfl_xor(sum, 16, 32);
    if (half == 0) rinv[r] = 1.0f / sum;
  }
  __syncthreads();

  // ctx: lane -> (q = lane>>1, d0 = (lane&1)*2)
  int qr = lane >> 1, d0 = (lane & 1) * 2;
  float a0 = 0.0f, a1 = 0.0f;
  for (int kk = 0; kk < SEQ; ++kk) {
    float w = sc[qr * SEQ + kk];
    a0 += w * vl[kk * 4 + d0];
    a1 += w * vl[kk * 4 + d0 + 1];
  }
  float inv = rinv[qr];
  size_t o = ((size_t)(b * SEQ + qt * 16 + qr)) * DIM + hh * 4 + d0;
  ctx[o]     = a0 * inv;
  ctx[o + 1] = a1 * inv;
}

// ---------------------------------------------------------------------------
// Output projection + residual + LayerNorm.
// ---------------------------------------------------------------------------
__global__ void proj_ln_kernel(const float* __restrict__ in,
                               const float* __restrict__ W,
                               const float* __restrict__ bias,
                               float* __restrict__ h,
                               const float* __restrict__ lng,
                               const float* __restrict__ lnb) {
  __shared__ float alds[16 * DIM];  // 7 KB A strip
  __shared__ float vbuf[16 * DIM];
  int mt = blockIdx.x;
  int tid = threadIdx.x;
  int wave = tid >> 5, lane = tid & 31;
  int half = lane >> 4, l = lane & 15;

  const float* Ag = in + ((size_t)mt * 16) * DIM;
#if defined(HAS_TDM)
  if (wave == 0) {
    tdm_load_2d(Ag, (unsigned)(size_t)(void*)alds, DIM, NTOK, DIM, 16, DIM);
    tdm_wait();
  }
#else
  for (int i = tid; i < 16 * DIM; i += 256) alds[i] = Ag[i];
#endif
  __syncthreads();

  if (wave < 7) {
    int n0 = wave * 16;
    const float* Bb = W + (size_t)n0 * DIM;
    v8f c = {};
#pragma unroll
    for (int kc = 0; kc < 28; ++kc)
      c = wmma4(load_frag(alds + kc * 4, DIM), load_frag(Bb + kc * 4, DIM), c);
    int col = n0 + l;
    float bval = bias[col];
#pragma unroll
    for (int j = 0; j < 8; ++j) {
      int m = j + 8 * half;
      vbuf[m * DIM + col] =
          c[j] + bval + h[((size_t)mt * 16 + m) * DIM + col];
    }
  }
  __syncthreads();

  int row = tid >> 4, gl = tid & 15;
  float s = 0.0f, sq = 0.0f;
  float vals[7];
#pragma unroll
  for (int i = 0; i < 7; ++i) {
    float x = vbuf[row * DIM + gl + 16 * i];
    vals[i] = x; s += x; sq += x * x;
  }
#pragma unroll
  for (int off = 1; off <= 8; off <<= 1) {
    s  += __shfl_xor(s, off, 32);
    sq += __shfl_xor(sq, off, 32);
  }
  float mu  = s * (1.0f / 112.0f);
  float var = sq * (1.0f / 112.0f) - mu * mu;
  float rs  = rsqrtf(var + 1e-5f);
#pragma unroll
  for (int i = 0; i < 7; ++i) {
    int col = gl + 16 * i;
    h[((size_t)mt * 16 + row) * DIM + col] =
        (vals[i] - mu) * rs * lng[col] + lnb[col];
  }
}

// ---------------------------------------------------------------------------
// KAN activations: SiLU(h) and cubic B-spline bases (Cox-de Boor).
// ---------------------------------------------------------------------------
__global__ void kan_act_kernel(const float* __restrict__ h,
                               const float* __restrict__ grid,
                               float* __restrict__ sb,
                               float* __restrict__ bsp) {
  int idx = blockIdx.x * blockDim.x + threadIdx.x;
  if (idx >= NTOK * DIM) return;
  float x = h[idx];
  sb[idx] = x / (1.0f + __expf(-x));

  float gg[12];
#pragma unroll
  for (int j = 0; j < 12; ++j) gg[j] = grid[j];   // all D rows identical
  float bb[11];
#pragma unroll
  for (int j = 0; j < 11; ++j)
    bb[j] = (x >= gg[j] && x < gg[j + 1]) ? 1.0f : 0.0f;
#pragma unroll
  for (int kk = 1; kk <= 3; ++kk) {
#pragma unroll
    for (int j = 0; j < 10; ++j) {
      if (j < 11 - kk) {
        bb[j] = (x - gg[j]) / (gg[j + kk] - gg[j]) * bb[j] +
                (gg[j + kk + 1] - x) / (gg[j + kk + 1] - gg[j + 1]) * bb[j + 1];
      }
    }
  }
  int n = idx / DIM, d = idx % DIM;
  size_t o = (size_t)n * SPW + d * NB8;
#pragma unroll
  for (int t = 0; t < NB8; ++t) bsp[o + t] = bb[t];
}

// sw[o][d*8+t] = spline_w[l,o,d,t] * scaler[l,o,d]
__global__ void sw_prep_kernel(const float* __restrict__ spw,
                               const float* __restrict__ scl,
                               float* __restrict__ sw) {
  int i = blockIdx.x * blockDim.x + threadIdx.x;
  const int total = 2 * DIM * DIM * NB8;
  if (i >= total) return;
  int d  = (i >> 3) % DIM;
  int od = i / (NB8 * DIM);           // l*DIM + o
  sw[i] = spw[i] * scl[od * DIM + d];
}

// ---------------------------------------------------------------------------
// KAN linear (base + spline GEMMs) + residual + LayerNorm2.  TDM stages the
// 16x896 bsp strip (56 KB) into LDS; 252 WMMAs per wave.
// ---------------------------------------------------------------------------
__global__ void kan_ln_kernel(const float* __restrict__ sb,
                              const float* __restrict__ bsp,
                              const float* __restrict__ bw,
                              const float* __restrict__ sw,
                              float* __restrict__ h,
                              const float* __restrict__ lng,
                              const float* __restrict__ lnb) {
  __shared__ float alds2[16 * SPW]; // 56 KB bsp strip
  __shared__ float vbuf[16 * DIM];  // 7 KB
  int mt = blockIdx.x;
  int tid = threadIdx.x;
  int wave = tid >> 5, lane = tid & 31;
  int half = lane >> 4, l = lane & 15;

  const float* A2g = bsp + ((size_t)mt * 16) * SPW;
#if defined(HAS_TDM)
  if (wave == 0) {
    tdm_load_2d(A2g, (unsigned)(size_t)(void*)alds2, SPW, NTOK, SPW, 16, SPW);
    tdm_wait();
  }
#else
  for (int i = tid; i < 16 * SPW; i += 256) alds2[i] = A2g[i];
#endif
  __syncthreads();

  if (wave < 7) {
    int n0 = wave * 16;
    v8f c = {};
    const float* Ab = sb + ((size_t)mt * 16) * DIM;
    const float* Bb = bw + (size_t)n0 * DIM;
#pragma unroll
    for (int kc = 0; kc < 28; ++kc)
      c = wmma4(load_frag(Ab + kc * 4, DIM), load_frag(Bb + kc * 4, DIM), c);
    const float* Bb2 = sw + (size_t)n0 * SPW;
#pragma unroll 8
    for (int kc = 0; kc < 224; ++kc)
      c = wmma4(load_frag(alds2 + kc * 4, SPW), load_frag(Bb2 + kc * 4, SPW), c);
    int col = n0 + l;
#pragma unroll
    for (int j = 0; j < 8; ++j) {
      int m = j + 8 * half;
      vbuf[m * DIM + col] = c[j] + h[((size_t)mt * 16 + m) * DIM + col];
    }
  }
  __syncthreads();

  int row = tid >> 4, gl = tid & 15;
  float s = 0.0f, sq = 0.0f;
  float vals[7];
#pragma unroll
  for (int i = 0; i < 7; ++i) {
    float x = vbuf[row * DIM + gl + 16 * i];
    vals[i] = x; s += x; sq += x * x;
  }
#pragma unroll
  for (int off = 1; off <= 8; off <<= 1) {
    s  += __shfl_xor(s, off, 32);
    sq += __shfl_xor(sq, off, 32);
  }
  float mu  = s * (1.0f / 112.0f);
  float var = sq * (1.0f / 112.0f) - mu * mu;
  float rs  = rsqrtf(var + 1e-5f);
#pragma unroll
  for (int i = 0; i < 7; ++i) {
    int col = gl + 16 * i;
    h[((size_t)mt * 16 + row) * DIM + col] =
        (vals[i] - mu) * rs * lng[col] + lnb[col];
  }
}

// Classifier: 32 lanes = 16 batches x 2 classes.
__global__ void cls_kernel(const float* __restrict__ h,
                           const float* __restrict__ cw,
                           const float* __restrict__ cb,
                           float* __restrict__ out) {
  int t = threadIdx.x;
  if (t >= 32) return;
  int b = t >> 1, c = t & 1;
  float acc = cb[c];
  const float* hr = h + (size_t)b * SEQ * DIM;   // token 0 of batch b
  const float* wr = cw + c * DIM;
  for (int i = 0; i < DIM; ++i) acc += hr[i] * wr[i];
  out[t] = acc;
}

// ---------------------------------------------------------------------------
extern "C" void kernel_launch(void* const* d_in, const int* in_sizes, int n_in,
                              void* d_out, int out_size, void* d_ws, size_t ws_size,
                              hipStream_t stream) {
  const float* x      = (const float*)d_in[0];
  const int*   amask  = (const int*)d_in[1];
  const float* grid   = (const float*)d_in[2];
  const float* qw     = (const float*)d_in[3];
  const float* qb     = (const float*)d_in[4];
  const float* kw     = (const float*)d_in[5];
  const float* kb     = (const float*)d_in[6];
  const float* vw     = (const float*)d_in[7];
  const float* vb     = (const float*)d_in[8];
  const float* ow     = (const float*)d_in[9];
  const float* ob     = (const float*)d_in[10];
  const float* ln1g   = (const float*)d_in[11];
  const float* ln1b   = (const float*)d_in[12];
  const float* ln2g   = (const float*)d_in[13];
  const float* ln2b   = (const float*)d_in[14];
  const float* basew  = (const float*)d_in[15];
  const float* splw   = (const float*)d_in[16];
  const float* splsc  = (const float*)d_in[17];
  const float* clsw   = (const float*)d_in[18];
  const float* clsb   = (const float*)d_in[19];
  float* out = (float*)d_out;

  const size_t ACT = (size_t)NTOK * DIM * sizeof(float);   // 3,670,016 B
  char* ws = (char*)d_ws;
  float* h    = (float*)(ws);
  float* q    = (float*)(ws + ACT);
  float* k    = (float*)(ws + 2 * ACT);
  float* v    = (float*)(ws + 3 * ACT);
  float* ctx  = (float*)(ws + 4 * ACT);
  float* sb   = (float*)(ws + 5 * ACT);
  float* sw   = (float*)(ws + 6 * ACT);                    // 802,816 B
  float* bsp  = (float*)(ws + 6 * ACT + 1024 * 1024);      // 29,360,128 B

  hipMemcpyAsync(h, x, ACT, hipMemcpyDeviceToDevice, stream);

  {
    int total = 2 * DIM * DIM * NB8;
    sw_prep_kernel<<<(total + 255) / 256, 256, 0, stream>>>(splw, splsc, sw);
  }

  const int DD = DIM * DIM;
  for (int l = 0; l < 2; ++l) {
    qkv_kernel<<<NTOK / 16, 256, 0, stream>>>(
        h, qw + l * DD, qb + l * DIM, kw + l * DD, kb + l * DIM,
        vw + l * DD, vb + l * DIM, q, k, v);

    attn_kernel<<<dim3(SEQ / 16, 28, 16), 32, 0, stream>>>(q, k, v, amask, ctx);

    proj_ln_kernel<<<NTOK / 16, 256, 0, stream>>>(
        ctx, ow + l * DD, ob + l * DIM, h, ln1g + l * DIM, ln1b + l * DIM);

    kan_act_kernel<<<(NTOK * DIM + 255) / 256, 256, 0, stream>>>(h, grid, sb, bsp);

    kan_ln_kernel<<<NTOK / 16, 256, 0, stream>>>(
        sb, bsp, basew + l * DD, sw + l * DIM * SPW, h,
        ln2g + l * DIM, ln2b + l * DIM);
  }

  cls_kernel<<<1, 32, 0, stream>>>(h, clsw, clsb, out);
}